// ExpKernelAttention_82592221102307
// MI455X (gfx1250) — compile-verified
//
#include <hip/hip_runtime.h>

// Problem constants (B,H,L,D) = (4,16,2048,64), fp32 in/out.
#define BB 4
#define HH 16
#define LL 2048
#define DD 64

typedef __attribute__((ext_vector_type(2))) float v2f;
typedef __attribute__((ext_vector_type(8))) float v8f;

// log2(e) / (2*sqrt(D)) = 1.4426950408889634 / 16
#define L2E_OVER_SCALE  0.09016844005556021f
// 2 * log2(e) / 16 (folded "2*qk" factor)
#define TWO_L2E_OVER_SCALE 0.18033688011112042f

// One 16(j) x 16(i) score tile: C = K_tile * Q_tile^T via fp32 WMMA, K=64 in 16 chunks of 4.
// A-frag (16x4 f32): lane l -> row l%16, VGPR g holds K = 2*(l/16)+g  -> contiguous float2.
// B-frag (4x16 f32): lane l -> col l%16, VGPR g holds K = 2*(l/16)+g  -> contiguous float2.
__device__ __forceinline__ v8f qk_tile(const v2f (&akf)[16],
                                       const float* __restrict__ Qbh,
                                       int i0, int ln, int hl) {
  v8f c = {};
  const float* qb = Qbh + (size_t)(i0 + ln) * DD + 2 * hl;
#pragma unroll
  for (int f = 0; f < 16; ++f) {
    v2f bf = *(const v2f*)(qb + 4 * f);
    c = __builtin_amdgcn_wmma_f32_16x16x4_f32(false, akf[f], false, bf,
                                              (short)0, c, false, false);
  }
  return c;
}

__global__ __launch_bounds__(256) void expattn_kernel(
    const float* __restrict__ Q, const float* __restrict__ K,
    const float* __restrict__ V, const int* __restrict__ Msk,
    float* __restrict__ out, float* __restrict__ attn) {
  __shared__ float t_sh[LL];                       // 8 KB: per-column bias t[i]
  __shared__ __align__(16) float p_sh[8][16 * 18]; // 9 KB: per-wave tile transpose pad

  const int tid  = threadIdx.x;
  const int wave = tid >> 5;
  const int lane = tid & 31;
  const int ln   = lane & 15;  // column / row-within-half
  const int hl   = lane >> 4;  // half-wave select

  const int ntile = LL / 128;            // 16 j-tiles per (b,h)
  const int jt = blockIdx.x % ntile;
  const int bh = blockIdx.x / ntile;
  const int b  = bh / HH;

  const float* Qbh = Q + (size_t)bh * LL * DD;
  const float* Kbh = K + (size_t)bh * LL * DD;
  const float* Vbh = V + (size_t)bh * LL * DD;
  const int*   Mb  = Msk + (size_t)b * LL * LL;
  float* outbh  = out  + (size_t)bh * LL * DD;
  float* attnbh = attn + (size_t)bh * LL * LL;

  // ---- prologue: t[i] = ||k_i||^2 - ||q_i||^2 into LDS (Q/K are L2-resident) ----
  for (int i = tid; i < LL; i += 256) {
    const float4* qv = (const float4*)(Qbh + (size_t)i * DD);
    const float4* kv = (const float4*)(Kbh + (size_t)i * DD);
    float sq = 0.f, sk = 0.f;
#pragma unroll
    for (int u = 0; u < DD / 4; ++u) {
      float4 q = qv[u], k = kv[u];
      sq += q.x * q.x + q.y * q.y + q.z * q.z + q.w * q.w;
      sk += k.x * k.x + k.y * k.y + k.z * k.z + k.w * k.w;
    }
    t_sh[i] = sk - sq;
  }
  __syncthreads();

  const int j0 = jt * 128 + wave * 16;   // this wave's 16 score rows (key index j)

  // Cache the wave's 16 K-rows as fp32 WMMA A-fragments (32 VGPRs).
  v2f akf[16];
  {
    const float* kr = Kbh + (size_t)(j0 + ln) * DD + 2 * hl;
#pragma unroll
    for (int f = 0; f < 16; ++f) akf[f] = *(const v2f*)(kr + 4 * f);
  }

  float* psw = &p_sh[wave][0];
  const v2f ones = {1.0f, 1.0f};

  // ---- pass A: row sums of exp(s) via WMMA-with-ones (no shuffles, no max shift) ----
  // sumacc C-layout: VGPR r holds rowsum for row j0 + r + 8*hl, replicated over lanes.
  v8f sumacc = {};
  for (int it = 0; it < LL / 16; ++it) {
    const int i0 = it * 16;
    v8f c = qk_tile(akf, Qbh, i0, ln, hl);
    const float tcs = t_sh[i0 + ln] * L2E_OVER_SCALE;
#pragma unroll
    for (int r = 0; r < 8; ++r) {
      const int row = j0 + r + 8 * hl;
      const int mv  = Mb[(size_t)row * LL + i0 + ln];
      const float e = (mv == 0) ? 0.f
                                : exp2f(fmaf(c[r], TWO_L2E_OVER_SCALE, tcs));
      psw[(r + 8 * hl) * 18 + ln] = e;   // E tile in (row, col) form
    }
    // sumacc += E(16j x 4i) * Ones(4 x 16): row sums land in C layout by row.
#pragma unroll
    for (int f = 0; f < 4; ++f) {
      v2f ae = *(const v2f*)(psw + ln * 18 + 4 * f + 2 * hl);
      sumacc = __builtin_amdgcn_wmma_f32_16x16x4_f32(false, ae, false, ones,
                                                     (short)0, sumacc, false, false);
    }
  }

  float inv[8];
#pragma unroll
  for (int r = 0; r < 8; ++r) inv[r] = 1.0f / sumacc[r];

  // ---- pass B: write normalized attn (1 GB NT stream) + PV via WMMA ----
  v8f acc[4];
  {
    v8f z = {};
#pragma unroll
    for (int n = 0; n < 4; ++n) acc[n] = z;
  }

  for (int it = 0; it < LL / 16; ++it) {
    const int i0 = it * 16;
    v8f c = qk_tile(akf, Qbh, i0, ln, hl);
    const float tcs = t_sh[i0 + ln] * L2E_OVER_SCALE;
#pragma unroll
    for (int r = 0; r < 8; ++r) {
      const int row = j0 + r + 8 * hl;
      const int mv  = Mb[(size_t)row * LL + i0 + ln];
      const float p = (mv == 0) ? 0.f
                                : exp2f(fmaf(c[r], TWO_L2E_OVER_SCALE, tcs)) * inv[r];
      __builtin_nontemporal_store(p, attnbh + (size_t)row * LL + i0 + ln);
      psw[(r + 8 * hl) * 18 + ln] = p;   // normalized P tile for re-fragmentation
    }
    // PV: Out(16j x 64d) += P(16j x 16i) * V(16i x 64d), K in chunks of 4.
#pragma unroll
    for (int f = 0; f < 4; ++f) {
      v2f ap = *(const v2f*)(psw + ln * 18 + 4 * f + 2 * hl);  // A: P[j=ln][i=4f+2hl+g]
      const float* vr = Vbh + (size_t)(i0 + 4 * f + 2 * hl) * DD + ln;
#pragma unroll
      for (int n = 0; n < 4; ++n) {
        v2f bv;
        bv[0] = vr[16 * n];        // B: V[i=4f+2hl  ][d=16n+ln]
        bv[1] = vr[16 * n + DD];   //    V[i=4f+2hl+1][d=16n+ln]
        acc[n] = __builtin_amdgcn_wmma_f32_16x16x4_f32(false, ap, false, bv,
                                                       (short)0, acc[n], false, false);
      }
    }
  }

  // ---- epilogue: out[b,h,j,d] ----
#pragma unroll
  for (int n = 0; n < 4; ++n)
#pragma unroll
    for (int r = 0; r < 8; ++r)
      outbh[(size_t)(j0 + r + 8 * hl) * DD + 16 * n + ln] = acc[n][r];
}

extern "C" void kernel_launch(void* const* d_in, const int* in_sizes, int n_in,
                              void* d_out, int out_size, void* d_ws, size_t ws_size,
                              hipStream_t stream) {
  (void)in_sizes; (void)n_in; (void)out_size; (void)d_ws; (void)ws_size;
  const float* Q = (const float*)d_in[0];
  const float* K = (const float*)d_in[1];
  const float* V = (const float*)d_in[2];
  const int*   M = (const int*)d_in[3];
  float* out  = (float*)d_out;
  float* attn = out + (size_t)BB * HH * LL * DD;  // d_out = [out | attn]

  dim3 grid(BB * HH * (LL / 128));  // 1024 workgroups, one per (b,h,j-tile)
  expattn_kernel<<<grid, 256, 0, stream>>>(Q, K, V, M, out, attn);
}